// Net_41420664602632
// MI455X (gfx1250) — compile-verified
//
#include <hip/hip_runtime.h>

// Leave-one-out Nadaraya-Watson kernel regression for MI455X (gfx1250).
// exp() dominates (3*N^2 = 50M transcendentals); all data fits in LDS.
// Each wave owns 16 query points; V_WMMA_F32_16X16X4_F32 folds both reductions
// (numerator = sum_j K*Y, denominator = sum_j K) into one matrix op per channel:
//   A(16x4): row0 = Y[j0..j3,c], row1 = 1,1,1,1, rows2..15 = 0
//   B(4x16): K[j0..j3, i0..i15] for channel c
//   D row0 accumulates numerator, row1 accumulates denominator.
// Projections are pre-scaled by s = sqrt(0.5*log2(e))/h so that
//   K = exp(-0.5*((Ft-Fx)/h)^2) = exp2(-(Ft'-Fx')^2) = exp2(d' * -d')
// costing sub + mul(with free neg modifier) + v_exp_f32 per element.
// Leave-one-out handled OUTSIDE the hot loop: at j==i, d'==0 so K==1.0,
// contributing exactly Y[i,c] / 1.0 to num/den; subtracted once at the end.

typedef __attribute__((ext_vector_type(2))) float v2f;
typedef __attribute__((ext_vector_type(8))) float v8f;

#define N_PTS         4096
#define J_TILE        512
#define BLOCK_THREADS 256
#define I_PER_WAVE    16
#define I_PER_BLOCK   128   // 8 waves * 16
#define JSPLIT        8

// sqrt(0.5 * log2(e))
#define SQRT_HALF_LOG2E 0.84932180028801904272f

__device__ __forceinline__ float fast_exp2(float v) {
#if __has_builtin(__builtin_amdgcn_exp2f)
    return __builtin_amdgcn_exp2f(v);
#else
    return exp2f(v);
#endif
}

template <bool DIRECT>
__global__ __launch_bounds__(BLOCK_THREADS)
void kde_loo_kernel(const float* __restrict__ x,
                    const float* __restrict__ train_X,
                    const float* __restrict__ Y,
                    const float* __restrict__ W,
                    const float* __restrict__ h,
                    float* __restrict__ out,
                    float* __restrict__ ws_num,
                    float* __restrict__ ws_den) {
    // Packed per-j record (24B): [Ft'0, Ft'1, Ft'2, Y0, Y1, Y2].
    // A j-pair (even base) is 48B, 16B-aligned -> three ds_load_b128.
    __shared__ __align__(16) float lds[J_TILE * 6];

    const int tid  = threadIdx.x;
    const int lane = tid & 31;
    const int wave = tid >> 5;
    const int r    = lane & 15;   // query column within the wave's 16-i tile
    const int jsel = lane >> 4;   // 0: K rows {0,1}, 1: K rows {2,3}

    // fc1 weight (3x4), uniform -> scalar loads
    float w[12];
#pragma unroll
    for (int k = 0; k < 12; ++k) w[k] = W[k];

    const float s = SQRT_HALF_LOG2E / h[0];   // fold bandwidth + log2e into scale

    const int i_global = blockIdx.x * I_PER_BLOCK + wave * I_PER_WAVE + r;

    // Scaled Fx for this lane's query point (both half-waves replicate same i)
    const float4 xi = ((const float4*)x)[i_global];
    const float fxs0 = (w[0]*xi.x + w[1]*xi.y + w[2]*xi.z  + w[3]*xi.w)  * s;
    const float fxs1 = (w[4]*xi.x + w[5]*xi.y + w[6]*xi.z  + w[7]*xi.w)  * s;
    const float fxs2 = (w[8]*xi.x + w[9]*xi.y + w[10]*xi.z + w[11]*xi.w) * s;

    // A-matrix row masks: row0 carries Y, row1 carries ones -> a = fma(y,m0,m1)
    const float m0 = (r == 0) ? 1.0f : 0.0f;
    const float m1 = (r == 1) ? 1.0f : 0.0f;

    v8f acc0 = {}; v8f acc1 = {}; v8f acc2 = {};

    const int jlo = DIRECT ? 0     : blockIdx.y * (N_PTS / JSPLIT);
    const int jhi = DIRECT ? N_PTS : jlo + (N_PTS / JSPLIT);

    for (int jt = jlo; jt < jhi; jt += J_TILE) {
        __syncthreads();
        // Stage scaled Ft = (train_X @ W^T)*s and Y for this j-tile into LDS.
        for (int sj = tid; sj < J_TILE; sj += BLOCK_THREADS) {
            const int j = jt + sj;
            const float4 tj = ((const float4*)train_X)[j];
            lds[sj*6+0] = (w[0]*tj.x + w[1]*tj.y + w[2]*tj.z  + w[3]*tj.w)  * s;
            lds[sj*6+1] = (w[4]*tj.x + w[5]*tj.y + w[6]*tj.z  + w[7]*tj.w)  * s;
            lds[sj*6+2] = (w[8]*tj.x + w[9]*tj.y + w[10]*tj.z + w[11]*tj.w) * s;
            lds[sj*6+3] = Y[j*3+0];
            lds[sj*6+4] = Y[j*3+1];
            lds[sj*6+5] = Y[j*3+2];
        }
        __syncthreads();

#pragma unroll 4
        for (int jj = 0; jj < J_TILE; jj += 4) {
            const int sA = jj + 2*jsel;        // even -> 48B-aligned record pair
            const float* base = &lds[sA*6];
            const float4 p0 = *(const float4*)(base);      // ftA0 ftA1 ftA2 yA0
            const float4 p1 = *(const float4*)(base + 4);  // yA1  yA2  ftB0 ftB1
            const float4 p2 = *(const float4*)(base + 8);  // ftB2 yB0  yB1  yB2

            { // channel 0
                const float dA = p0.x - fxs0, dB = p1.z - fxs0;
                v2f B = { fast_exp2(dA * -dA), fast_exp2(dB * -dB) };
                v2f A = { fmaf(p0.w, m0, m1), fmaf(p2.y, m0, m1) };
                acc0 = __builtin_amdgcn_wmma_f32_16x16x4_f32(
                           false, A, false, B, (short)0, acc0, false, false);
            }
            { // channel 1
                const float dA = p0.y - fxs1, dB = p1.w - fxs1;
                v2f B = { fast_exp2(dA * -dA), fast_exp2(dB * -dB) };
                v2f A = { fmaf(p1.x, m0, m1), fmaf(p2.z, m0, m1) };
                acc1 = __builtin_amdgcn_wmma_f32_16x16x4_f32(
                           false, A, false, B, (short)0, acc1, false, false);
            }
            { // channel 2
                const float dA = p0.z - fxs2, dB = p2.x - fxs2;
                v2f B = { fast_exp2(dA * -dA), fast_exp2(dB * -dB) };
                v2f A = { fmaf(p1.y, m0, m1), fmaf(p2.w, m0, m1) };
                acc2 = __builtin_amdgcn_wmma_f32_16x16x4_f32(
                           false, A, false, B, (short)0, acc2, false, false);
            }
        }
    }

    // D layout (32-bit C/D, 16x16): VGPR0 lanes0-15 = row M=0 (numerator),
    // VGPR1 lanes0-15 = row M=1 (denominator).
    if (lane < 16) {
        const float n0 = acc0[0], d0 = acc0[1];
        const float n1 = acc1[0], d1 = acc1[1];
        const float n2 = acc2[0], d2 = acc2[1];
        if (DIRECT) {
            // subtract the exact self-pair contribution (K_self == 1.0)
            const float y0 = Y[i_global*3+0];
            const float y1 = Y[i_global*3+1];
            const float y2 = Y[i_global*3+2];
            out[i_global*3+0] = (n0 - y0) / (d0 - 1.0f);
            out[i_global*3+1] = (n1 - y1) / (d1 - 1.0f);
            out[i_global*3+2] = (n2 - y2) / (d2 - 1.0f);
        } else {
            atomicAdd(&ws_num[i_global*3+0], n0);
            atomicAdd(&ws_num[i_global*3+1], n1);
            atomicAdd(&ws_num[i_global*3+2], n2);
            atomicAdd(&ws_den[i_global*3+0], d0);
            atomicAdd(&ws_den[i_global*3+1], d1);
            atomicAdd(&ws_den[i_global*3+2], d2);
        }
    }
}

__global__ __launch_bounds__(256)
void kde_finalize(const float* __restrict__ num,
                  const float* __restrict__ den,
                  const float* __restrict__ Y,
                  float* __restrict__ out, int n) {
    const int idx = blockIdx.x * blockDim.x + threadIdx.x;
    if (idx < n) {
        // full sums accumulated over all j; remove the exact self-pair term
        out[idx] = (num[idx] - Y[idx]) / (den[idx] - 1.0f);
    }
}

extern "C" void kernel_launch(void* const* d_in, const int* in_sizes, int n_in,
                              void* d_out, int out_size, void* d_ws, size_t ws_size,
                              hipStream_t stream) {
    const float* x       = (const float*)d_in[0];
    const float* train_X = (const float*)d_in[1];
    const float* Y       = (const float*)d_in[2];
    const float* W       = (const float*)d_in[3];
    const float* h       = (const float*)d_in[4];
    float* out = (float*)d_out;

    const size_t need = (size_t)2 * N_PTS * 3 * sizeof(float); // num + den partials

    if (ws_size >= need) {
        float* ws_num = (float*)d_ws;
        float* ws_den = ws_num + N_PTS * 3;
        hipMemsetAsync(d_ws, 0, need, stream);
        dim3 grid(N_PTS / I_PER_BLOCK, JSPLIT);
        kde_loo_kernel<false><<<grid, BLOCK_THREADS, 0, stream>>>(
            x, train_X, Y, W, h, out, ws_num, ws_den);
        kde_finalize<<<(N_PTS * 3 + 255) / 256, 256, 0, stream>>>(
            ws_num, ws_den, Y, out, N_PTS * 3);
    } else {
        dim3 grid(N_PTS / I_PER_BLOCK, 1);
        kde_loo_kernel<true><<<grid, BLOCK_THREADS, 0, stream>>>(
            x, train_X, Y, W, h, out, nullptr, nullptr);
    }
}